// Model_21629455302825
// MI455X (gfx1250) — compile-verified
//
#include <hip/hip_runtime.h>

// ---------------------------------------------------------------------------
// Segmented LoRA expand for MI455X (gfx1250), fp32 end-to-end via
// V_WMMA_F32_16X16X4_F32 (wave32 WMMA).
// Roofline: 268 MB mandatory output stores @23.3 TB/s ~= 11.5 us floor vs
// 5.4 GFLOP fp32 -> keep reference fp32 precision, stream stores NT.
// Round-2 fixes: rank dispatched once per block (templated NK4), B loads
// clause-batched before the WMMA chain, dual accumulator chains for ILP,
// output row pointers hoisted out of the tile loop.
// ---------------------------------------------------------------------------

typedef float v2f __attribute__((ext_vector_type(2)));
typedef float v8f __attribute__((ext_vector_type(8)));

constexpr int MAX_RANK   = 64;
constexpr int NUM_SLICES = 2;
constexpr int IN_COLS    = NUM_SLICES * MAX_RANK;  // 128
constexpr int OUT_DIM    = 4096;
constexpr int LDS_STRIDE = IN_COLS + 4;            // 132: banks = (4m + c) % 64, conflict-free
constexpr int TILE_ROWS  = 16;
constexpr int WAVES_PER_SLICE = 4;                 // 8 waves total, 2 slices
constexpr int YSPLIT     = 4;                      // column-sweep split for parallelism

// Sweep this wave's 16-wide column tiles with a compile-time K-step count.
// NK4 is ceil(r/4) rounded up to a multiple of 4; correctness for any r comes
// from zero-padded A fragments (k >= r) and always-valid W memory (k < 64).
template <int NK4>
__device__ __forceinline__ void sweep_tiles(const v2f (&afrag)[16],
                                            const float* __restrict__ wbase,
                                            float* const (&rowptr)[8],
                                            int so_j, int tiles, int t0, int tstride,
                                            int laneN, int laneH, float scal)
{
    for (int t = t0; t < tiles; t += tstride) {
        const int n = so_j + t * 16 + laneN;
        const float* wp = wbase + (size_t)n * MAX_RANK + 2 * laneH;

        // Batch all B fragments: one base address, immediate offsets -> load clause.
        v2f bfrag[NK4 > 0 ? NK4 : 1];
#pragma unroll
        for (int f = 0; f < NK4; ++f) {
            const float2 bv = *(const float2*)(wp + 4 * f);
            bfrag[f][0] = bv.x;
            bfrag[f][1] = bv.y;
        }

        // Two independent accumulator chains to break the WMMA RAW chain.
        v8f c0 = {0.f, 0.f, 0.f, 0.f, 0.f, 0.f, 0.f, 0.f};
        v8f c1 = {0.f, 0.f, 0.f, 0.f, 0.f, 0.f, 0.f, 0.f};
#pragma unroll
        for (int f = 0; f + 1 < NK4; f += 2) {
            c0 = __builtin_amdgcn_wmma_f32_16x16x4_f32(
                     false, afrag[f],     false, bfrag[f],     (short)0, c0, false, false);
            c1 = __builtin_amdgcn_wmma_f32_16x16x4_f32(
                     false, afrag[f + 1], false, bfrag[f + 1], (short)0, c1, false, false);
        }
        if (NK4 & 1) {
            c0 = __builtin_amdgcn_wmma_f32_16x16x4_f32(
                     false, afrag[NK4 - 1], false, bfrag[NK4 - 1], (short)0, c0, false, false);
        }

        // D layout: VGPR v -> M = v + 8*laneH, N = laneN. Rows scattered by
        // permutation (rowptr precomputed); lanes 0-15 give 64B-contiguous runs.
#pragma unroll
        for (int v = 0; v < 8; ++v) {
            __builtin_nontemporal_store((c0[v] + c1[v]) * scal, rowptr[v] + n);
        }
    }
}

__global__ __launch_bounds__(256)
void lora_expand_wmma_f32(const float* __restrict__ x,
                          const float* __restrict__ weights,
                          const int*   __restrict__ seg_indptr,
                          const int*   __restrict__ weight_indices,
                          const int*   __restrict__ lora_ranks,
                          const float* __restrict__ scalings,
                          const int*   __restrict__ permutation,
                          const int*   __restrict__ slice_offsets,
                          float*       __restrict__ out,
                          int num_segments)
{
    __shared__ __align__(16) float lds_x[TILE_ROWS * LDS_STRIDE];
    __shared__ int lds_perm[TILE_ROWS];

    const int tid      = threadIdx.x;
    const int row_base = blockIdx.x * TILE_ROWS;

    // ---- segment metadata (uniform across block; tiles never straddle segs) ----
    int s = 0;
    for (int i = 1; i < num_segments; ++i)
        if (row_base >= seg_indptr[i]) s = i;
    const int   w    = weight_indices[s];
    const int   r    = lora_ranks[w];          // 0 => zero output rows
    const float scal = scalings[w];

    // ---- stage permutation + 16 gathered x rows into LDS ----
    if (tid < TILE_ROWS) lds_perm[tid] = permutation[row_base + tid];
    {
        const int m  = tid >> 4;               // row within tile (16 threads/row)
        const int cg = (tid & 15) * 8;         // 8 floats per thread
        const int p  = permutation[row_base + m];
        const float4 a = *(const float4*)(x + (size_t)p * IN_COLS + cg);
        const float4 b = *(const float4*)(x + (size_t)p * IN_COLS + cg + 4);
        *(float4*)(&lds_x[m * LDS_STRIDE + cg])     = a;
        *(float4*)(&lds_x[m * LDS_STRIDE + cg + 4]) = b;
    }
    __syncthreads();

    const int wave  = tid >> 5;
    const int lane  = tid & 31;
    const int laneN = lane & 15;               // N / M index within half-wave
    const int laneH = lane >> 4;               // half-wave select

    // ---- A fragments: 16x4 f32 layout (lane L: M=L%16; VGPR v: K=4f+2*laneH+v) ----
    const int j     = wave / WAVES_PER_SLICE;  // slice of this wave
    const int xbase = j * r;                   // x packs slice j at columns [j*r, j*r+r)
    v2f afrag[16];
#pragma unroll
    for (int f = 0; f < 16; ++f) {
        const int k0 = 4 * f + 2 * laneH;
        float a0 = 0.0f, a1 = 0.0f;
        if (k0     < r) a0 = lds_x[laneN * LDS_STRIDE + xbase + k0];
        if (k0 + 1 < r) a1 = lds_x[laneN * LDS_STRIDE + xbase + k0 + 1];
        afrag[f][0] = a0;
        afrag[f][1] = a1;
    }

    // ---- hoist permuted output-row base pointers (invariant over tiles) ----
    float* rowptr[8];
#pragma unroll
    for (int v = 0; v < 8; ++v)
        rowptr[v] = out + (size_t)lds_perm[v + 8 * laneH] * OUT_DIM;

    // ---- sweep column tiles of this slice, rank resolved ONCE per block ----
    const int so_j    = slice_offsets[j];
    const int tiles   = (slice_offsets[j + 1] - so_j) >> 4;   // 128 per slice
    const int t0      = (wave % WAVES_PER_SLICE) + WAVES_PER_SLICE * blockIdx.y;
    const int tstride = WAVES_PER_SLICE * YSPLIT;
    const float* wbase = weights + (size_t)w * OUT_DIM * MAX_RANK;

    const int nk4 = (r + 3) >> 2;              // K-steps of 4; uniform branch below
    if (nk4 == 0) {
        sweep_tiles<0>(afrag, wbase, rowptr, so_j, tiles, t0, tstride, laneN, laneH, scal);
    } else if (nk4 <= 4) {
        sweep_tiles<4>(afrag, wbase, rowptr, so_j, tiles, t0, tstride, laneN, laneH, scal);
    } else if (nk4 <= 8) {
        sweep_tiles<8>(afrag, wbase, rowptr, so_j, tiles, t0, tstride, laneN, laneH, scal);
    } else if (nk4 <= 12) {
        sweep_tiles<12>(afrag, wbase, rowptr, so_j, tiles, t0, tstride, laneN, laneH, scal);
    } else {
        sweep_tiles<16>(afrag, wbase, rowptr, so_j, tiles, t0, tstride, laneN, laneH, scal);
    }
}

extern "C" void kernel_launch(void* const* d_in, const int* in_sizes, int n_in,
                              void* d_out, int out_size, void* d_ws, size_t ws_size,
                              hipStream_t stream) {
    const float* x              = (const float*)d_in[0];
    const float* weights        = (const float*)d_in[1];
    const int*   seg_indptr     = (const int*)d_in[2];
    const int*   weight_indices = (const int*)d_in[3];
    const int*   lora_ranks     = (const int*)d_in[4];
    const float* scalings       = (const float*)d_in[5];
    const int*   permutation    = (const int*)d_in[6];
    const int*   slice_offsets  = (const int*)d_in[7];

    const int M            = in_sizes[6];          // permutation has M entries
    const int num_segments = in_sizes[2] - 1;      // seg_indptr has NS+1 entries

    dim3 grid(M / TILE_ROWS, YSPLIT);
    lora_expand_wmma_f32<<<grid, 256, 0, stream>>>(
        x, weights, seg_indptr, weight_indices, lora_ranks, scalings,
        permutation, slice_offsets, (float*)d_out, num_segments);
}